// Decoder_15702400434891
// MI455X (gfx1250) — compile-verified
//
#include <hip/hip_runtime.h>
#include <math.h>

typedef __attribute__((ext_vector_type(2))) float v2f;
typedef __attribute__((ext_vector_type(8))) float v8f;

__device__ __forceinline__ v8f wmma_f32(v2f a, v2f b, v8f c) {
  return __builtin_amdgcn_wmma_f32_16x16x4_f32(false, a, false, b, (short)0, c, false, false);
}

// ---- constants of this problem ----
#define B_  64
#define S_  128
#define V_  32000
#define E_  256
#define EH_ 512
#define H_  512
#define A_  256

// K1: dec_proj[b][a] = ba[a] + hidden[b] . Wa_dec[:,a]      grid(64) block(256)
__global__ void k_dec_proj(const float* __restrict__ hidden,
                           const float* __restrict__ Wa_dec,
                           const float* __restrict__ ba,
                           float* __restrict__ dp) {
  int b = blockIdx.x, a = threadIdx.x;
  float s = ba[a];
  const float* hrow = hidden + b * H_;
  for (int h = 0; h < H_; ++h) s += hrow[h] * Wa_dec[h * A_ + a];
  dp[b * A_ + a] = s;
}

// K2: fused energy GEMM + tanh + v_att dot + mask -> scores
// rows = b*S+s flattened (8192), one 16-row tile per block; 8 waves x 2 col-tiles = 256 cols
// grid(512) block(256)
__global__ void k_attn_scores(const float* __restrict__ enc,
                              const float* __restrict__ Wa_enc,
                              const float* __restrict__ dp,
                              const float* __restrict__ v_att,
                              const unsigned char* __restrict__ mask,
                              float* __restrict__ scores) {
  const int lane = threadIdx.x & 31;
  const int wave = threadIdx.x >> 5;
  const int half = lane >> 4;
  const int lr   = lane & 15;
  const int mbase = blockIdx.x * 16;
  const int b = mbase >> 7;              // S_=128, 16-row tile never crosses b
  const int arow = mbase + lr;
  const int col0 = wave * 32 + lr;
  const int col1 = col0 + 16;

  __shared__ float s_part[16];
  if (threadIdx.x < 16) s_part[threadIdx.x] = 0.0f;
  __syncthreads();

  v8f acc0 = {}; v8f acc1 = {};
  const float* Arow = enc + (long)arow * EH_;
  for (int kb = 0; kb < EH_; kb += 4) {
    int k0 = kb + 2 * half;
    v2f a, b0, b1;
    a.x  = Arow[k0];                 a.y  = Arow[k0 + 1];
    b0.x = Wa_enc[k0 * A_ + col0];   b0.y = Wa_enc[(k0 + 1) * A_ + col0];
    b1.x = Wa_enc[k0 * A_ + col1];   b1.y = Wa_enc[(k0 + 1) * A_ + col1];
    acc0 = wmma_f32(a, b0, acc0);
    acc1 = wmma_f32(a, b1, acc1);
  }
  float va0 = v_att[col0], va1 = v_att[col1];
  float dp0 = dp[b * A_ + col0], dp1 = dp[b * A_ + col1];
  #pragma unroll
  for (int v = 0; v < 8; ++v) {
    int r = v + 8 * half;
    float e = tanhf(acc0[v] + dp0) * va0 + tanhf(acc1[v] + dp1) * va1;
    atomicAdd(&s_part[r], e);        // ds_add_f32
  }
  __syncthreads();
  if (threadIdx.x < 16) {
    int g = mbase + threadIdx.x;
    scores[g] = mask[g] ? s_part[threadIdx.x] : -1.0e9f;
  }
}

// K3: softmax over S per batch row -> attn (written to d_out attn section)  grid(64) block(128)
__global__ void k_softmax(const float* __restrict__ scores, float* __restrict__ attn) {
  __shared__ float red[S_];
  int b = blockIdx.x, s = threadIdx.x;
  float v = scores[b * S_ + s];
  red[s] = v; __syncthreads();
  for (int off = 64; off > 0; off >>= 1) { if (s < off) red[s] = fmaxf(red[s], red[s + off]); __syncthreads(); }
  float m = red[0]; __syncthreads();
  float e = expf(v - m);
  red[s] = e; __syncthreads();
  for (int off = 64; off > 0; off >>= 1) { if (s < off) red[s] += red[s + off]; __syncthreads(); }
  attn[b * S_ + s] = e / red[0];
}

// K4: context[b][h] = sum_s attn[b][s]*enc[b][s][h]      grid(64) block(256)
__global__ void k_context(const float* __restrict__ attn, const float* __restrict__ enc,
                          float* __restrict__ ctx) {
  __shared__ float w[S_];
  int b = blockIdx.x, t = threadIdx.x;
  if (t < S_) w[t] = attn[b * S_ + t];
  __syncthreads();
  for (int h = t; h < EH_; h += 256) {
    float s = 0.0f;
    const float* base = enc + (long)b * S_ * EH_ + h;
    for (int si = 0; si < S_; ++si) s += w[si] * base[si * EH_];
    ctx[b * EH_ + h] = s;
  }
}

// K5: embedded gather + x1 = [embedded | context]        grid(64) block(256)
__global__ void k_build_x1(const int* __restrict__ tok, const float* __restrict__ emb,
                           const float* __restrict__ ctx,
                           float* __restrict__ embedded, float* __restrict__ x1) {
  int b = blockIdx.x, t = threadIdx.x;
  float e = emb[(long)tok[b] * E_ + t];
  embedded[b * E_ + t] = e;
  x1[b * (E_ + EH_) + t] = e;
  x1[b * (E_ + EH_) + E_ + t]       = ctx[b * EH_ + t];
  x1[b * (E_ + EH_) + E_ + 256 + t] = ctx[b * EH_ + 256 + t];
}

// K6/K7: out[64 x N] = X[64 x K] @ W^T + bias,  W row-major [N x K]
// grid(N/128, 4) block(256): 8 waves, 1 col-tile each
__global__ void k_gemm_xwT(const float* __restrict__ X, const float* __restrict__ W,
                           const float* __restrict__ bias, float* __restrict__ out,
                           int N, int K) {
  const int lane = threadIdx.x & 31;
  const int wave = threadIdx.x >> 5;
  const int half = lane >> 4;
  const int lr   = lane & 15;
  const int mbase = blockIdx.y * 16;
  const int col = blockIdx.x * 128 + wave * 16 + lr;
  const float* Arow = X + (long)(mbase + lr) * K;
  const float* Wrow = W + (long)col * K;
  v8f acc = {};
  for (int kb = 0; kb < K; kb += 4) {
    int k0 = kb + 2 * half;
    v2f a, bb;
    a.x  = Arow[k0];  a.y  = Arow[k0 + 1];
    bb.x = Wrow[k0];  bb.y = Wrow[k0 + 1];
    acc = wmma_f32(a, bb, acc);
  }
  float bs = bias[col];
  #pragma unroll
  for (int v = 0; v < 8; ++v) {
    int row = mbase + v + 8 * half;
    out[(long)row * N + col] = acc[v] + bs;
  }
}

// K8: GRU gates + new_hidden + x2 = [new_hidden | context | embedded]   grid(64) block(256)
__global__ void k_gru(const float* __restrict__ gx, const float* __restrict__ gh,
                      const float* __restrict__ hidden, const float* __restrict__ ctx,
                      const float* __restrict__ embedded,
                      float* __restrict__ new_hidden, float* __restrict__ x2) {
  int b = blockIdx.x, t = threadIdx.x;
  const float* gxb = gx + b * 3 * H_;
  const float* ghb = gh + b * 3 * H_;
  for (int h = t; h < H_; h += 256) {
    float r = 1.0f / (1.0f + expf(-(gxb[h] + ghb[h])));
    float z = 1.0f / (1.0f + expf(-(gxb[H_ + h] + ghb[H_ + h])));
    float n = tanhf(gxb[2 * H_ + h] + r * ghb[2 * H_ + h]);
    float nh = (1.0f - z) * n + z * hidden[b * H_ + h];
    new_hidden[b * H_ + h] = nh;
    x2[b * 1280 + h] = nh;
    x2[b * 1280 + H_ + h] = ctx[b * EH_ + h];
  }
  x2[b * 1280 + 1024 + t] = embedded[b * E_ + t];
}

// K9: prediction[64 x V] = x2[64 x 1280] @ W_fc[1280 x V] + b_fc
// Memory-bound: stream W_fc exactly once. One block covers ALL 64 rows x 128 cols.
// W_fc staged through LDS in 16x128 chunks via async global->LDS b128, double buffered.
// grid(V/128) block(256): 8 waves, each 1 N-tile x 4 M-tiles.
#define PK_  1280
#define PKC  16                 // K rows per chunk
#define PNCH (PK_ / PKC)        // 80 chunks
__global__ void k_predict(const float* __restrict__ X, const float* __restrict__ W,
                          const float* __restrict__ bias, float* __restrict__ out) {
  const int N = V_;
  const int lane = threadIdx.x & 31;
  const int wave = threadIdx.x >> 5;
  const int half = lane >> 4;
  const int lr   = lane & 15;
  const int cbase = blockIdx.x * 128;
  const int coff  = wave * 16 + lr;          // column within block [0,128)

  __shared__ float buf[2][PKC * 128];        // 2 x 8KB

  // staging geometry: thread t copies 8 consecutive floats of the 16x128 chunk
  const int srow = threadIdx.x >> 4;         // 0..15
  const int scol = (threadIdx.x & 15) * 8;   // 0,8,...,120

  auto stage = [&](int c) {
    const float* g = W + (long)(c * PKC + srow) * N + cbase + scol;
    unsigned int l = (unsigned int)(unsigned long long)&buf[c & 1][srow * 128 + scol];
    // async copy: each op moves 16 bytes from global address g to LDS address l, same offset on both; tracked by ASYNCcnt
    asm volatile("global_load_async_to_lds_b128 %0, %1, off"
                 :: "v"(l), "v"(g) : "memory");
    asm volatile("global_load_async_to_lds_b128 %0, %1, off offset:16"
                 :: "v"(l), "v"(g) : "memory");
  };

  v8f acc[4] = {v8f{}, v8f{}, v8f{}, v8f{}};

  stage(0);
  for (int c = 0; c < PNCH; ++c) {
    if (c + 1 < PNCH) {
      stage(c + 1);
      // 2 ops of chunk c+1 outstanding; chunk c's ops (in-order) are complete
      asm volatile("s_wait_asynccnt 0x2" ::: "memory");
    } else {
      asm volatile("s_wait_asynccnt 0x0" ::: "memory");
    }
    __syncthreads();

    const float* bufc = buf[c & 1];
    const int kcg = c * PKC;
    #pragma unroll
    for (int ks = 0; ks < PKC / 4; ++ks) {
      const int k0l = ks * 4 + 2 * half;
      v2f bb;
      bb.x = bufc[k0l * 128 + coff];
      bb.y = bufc[(k0l + 1) * 128 + coff];
      const int kg = kcg + k0l;
      #pragma unroll
      for (int mt = 0; mt < 4; ++mt) {
        v2f a;
        const float* Arow = X + (long)(mt * 16 + lr) * PK_;
        a.x = Arow[kg];
        a.y = Arow[kg + 1];
        acc[mt] = wmma_f32(a, bb, acc[mt]);
      }
    }
    __syncthreads();   // safe to overwrite buf[c&1] two iterations later
  }

  const int col = cbase + coff;
  float bs = bias[col];
  #pragma unroll
  for (int mt = 0; mt < 4; ++mt) {
    #pragma unroll
    for (int v = 0; v < 8; ++v) {
      int row = mt * 16 + v + 8 * half;
      out[(long)row * N + col] = acc[mt][v] + bs;
    }
  }
}

extern "C" void kernel_launch(void* const* d_in, const int* in_sizes, int n_in,
                              void* d_out, int out_size, void* d_ws, size_t ws_size,
                              hipStream_t stream) {
  const int*   tok    = (const int*)d_in[0];
  const float* hidden = (const float*)d_in[1];
  const float* enc    = (const float*)d_in[2];
  const unsigned char* mask = (const unsigned char*)d_in[3];   // jnp.bool_ = 1 byte
  const float* emb    = (const float*)d_in[4];
  const float* Wa_enc = (const float*)d_in[5];
  const float* Wa_dec = (const float*)d_in[6];
  const float* ba     = (const float*)d_in[7];
  const float* v_att  = (const float*)d_in[8];
  const float* W_ih   = (const float*)d_in[9];
  const float* W_hh   = (const float*)d_in[10];
  const float* b_ih   = (const float*)d_in[11];
  const float* b_hh   = (const float*)d_in[12];
  const float* W_fc   = (const float*)d_in[13];
  const float* b_fc   = (const float*)d_in[14];

  float* out = (float*)d_out;
  float* prediction = out;                      // 64*32000
  float* new_hidden = out + (long)B_ * V_;      // 64*512
  float* attn       = new_hidden + B_ * H_;     // 64*128

  float* ws = (float*)d_ws;
  float* dp       = ws;                 // 64*256   = 16384
  float* scores   = dp + 16384;         // 64*128   = 8192
  float* ctx      = scores + 8192;      // 64*512   = 32768
  float* embedded = ctx + 32768;        // 64*256   = 16384
  float* x1       = embedded + 16384;   // 64*768   = 49152
  float* gx       = x1 + 49152;         // 64*1536  = 98304
  float* gh       = gx + 98304;         // 64*1536  = 98304
  float* x2       = gh + 98304;         // 64*1280  = 81920

  k_dec_proj   <<<B_, A_, 0, stream>>>(hidden, Wa_dec, ba, dp);
  k_attn_scores<<<(B_ * S_) / 16, 256, 0, stream>>>(enc, Wa_enc, dp, v_att, mask, scores);
  k_softmax    <<<B_, S_, 0, stream>>>(scores, attn);
  k_context    <<<B_, 256, 0, stream>>>(attn, enc, ctx);
  k_build_x1   <<<B_, 256, 0, stream>>>(tok, emb, ctx, embedded, x1);
  k_gemm_xwT   <<<dim3(1536 / 128, 4), 256, 0, stream>>>(x1, W_ih, b_ih, gx, 1536, 768);
  k_gemm_xwT   <<<dim3(1536 / 128, 4), 256, 0, stream>>>(hidden, W_hh, b_hh, gh, 1536, 512);
  k_gru        <<<B_, 256, 0, stream>>>(gx, gh, hidden, ctx, embedded, new_hidden, x2);
  k_predict    <<<V_ / 128, 256, 0, stream>>>(x2, W_fc, b_fc, prediction);
}